// SVDHead_no_network_16303695856096
// MI455X (gfx1250) — compile-verified
//
#include <hip/hip_runtime.h>
#include <hip/hip_bf16.h>
#include <math.h>

#define EPSC 1e-12f
#define ALPHA_FACTOR 4.0f
#define BIGF 1.0e30f
#define LOG2E 1.4426950408889634f

typedef __attribute__((ext_vector_type(2))) float v2f;
typedef __attribute__((ext_vector_type(8))) float v8f;

// D = A(16x4,f32) x B(4x16,f32) + C(16x16,f32)  -> v_wmma_f32_16x16x4_f32
__device__ __forceinline__ v8f wmma16x16x4(v2f a, v2f b, v8f c) {
  return __builtin_amdgcn_wmma_f32_16x16x4_f32(
      /*neg_a=*/false, a, /*neg_b=*/false, b,
      /*c_mod=*/(short)0, c, /*reuse_a=*/false, /*reuse_b=*/false);
}
// raw TRANS ops (v_sqrt_f32 / v_exp_f32 / v_rcp_f32), ~1ulp — no IEEE fixup code
__device__ __forceinline__ float fsqrt_(float x) { return __builtin_amdgcn_sqrtf(x); }
__device__ __forceinline__ float fexp2_(float x) { return __builtin_amdgcn_exp2f(x); }
__device__ __forceinline__ float frcp_(float x)  { return __builtin_amdgcn_rcpf(x); }

// ------------------------------------------------------------ self NN min --
// min_{m!=n} |src_n - src_m|^2 per src point, WMMA distance tiles.
// Diagonal intersects exactly one 16x16 tile (m0==n0): peel it so the main
// sweep carries no per-element masking.
__global__ void selfmin_kernel(const float* __restrict__ src,
                               float* __restrict__ nnmin, int N) {
  __shared__ float sx[2048], sy[2048], sz[2048];
  const int b = blockIdx.y;
  const float* S = src + (size_t)b * 3 * N;
  for (int i = threadIdx.x; i < N; i += blockDim.x) {
    sx[i] = S[i]; sy[i] = S[N + i]; sz[i] = S[2 * N + i];
  }
  __syncthreads();
  const int wv = threadIdx.x >> 5, lane = threadIdx.x & 31;
  const int hf = lane >> 4, nl = lane & 15;
  const int n0 = (blockIdx.x * 8 + wv) * 16;
  const int ar = n0 + nl;
  v2f A;
  if (hf == 0) { A.x = -2.f * sx[ar]; A.y = -2.f * sy[ar]; }
  else         { A.x = -2.f * sz[ar]; A.y = 0.f; }
  float sn2[8];
#pragma unroll
  for (int v = 0; v < 8; ++v) {
    int r = n0 + v + 8 * hf;
    float x = sx[r], y = sy[r], z = sz[r];
    sn2[v] = x * x + y * y + z * z;
  }
  float mn[8];
#pragma unroll
  for (int v = 0; v < 8; ++v) mn[v] = BIGF;

  // main sweep: all off-diagonal tiles, no masking (skip = 1 scalar branch)
  for (int m0 = 0; m0 < N; m0 += 16) {
    if (m0 == n0) continue;
    const int mc = m0 + nl;
    float tx = sx[mc], ty = sy[mc], tz = sz[mc];
    v2f Bf;
    if (hf == 0) { Bf.x = tx; Bf.y = ty; } else { Bf.x = tz; Bf.y = 0.f; }
    const float tm2 = tx * tx + ty * ty + tz * tz;
    v8f C;
#pragma unroll
    for (int v = 0; v < 8; ++v) C[v] = sn2[v] + tm2;
    C = wmma16x16x4(A, Bf, C);
#pragma unroll
    for (int v = 0; v < 8; ++v) mn[v] = fminf(mn[v], C[v]);
  }
  // peeled diagonal tile (m0 == n0): static per-v mask
  {
    const int mc = n0 + nl;
    float tx = sx[mc], ty = sy[mc], tz = sz[mc];
    v2f Bf;
    if (hf == 0) { Bf.x = tx; Bf.y = ty; } else { Bf.x = tz; Bf.y = 0.f; }
    const float tm2 = tx * tx + ty * ty + tz * tz;
    v8f C;
#pragma unroll
    for (int v = 0; v < 8; ++v) C[v] = sn2[v] + tm2;
    C = wmma16x16x4(A, Bf, C);
#pragma unroll
    for (int v = 0; v < 8; ++v) {
      float d2 = (nl == v + 8 * hf) ? BIGF : C[v];   // loop-invariant mask
      mn[v] = fminf(mn[v], d2);
    }
  }
#pragma unroll
  for (int v = 0; v < 8; ++v) {
    float x = mn[v];
    for (int off = 8; off > 0; off >>= 1) x = fminf(x, __shfl_xor(x, off));
    if (nl == 0) nnmin[(size_t)b * N + n0 + v + 8 * hf] = x;
  }
}

// t_b = ALPHA_FACTOR * mean(sqrt(max(nnmin,eps)))
__global__ void treduce_kernel(const float* __restrict__ nnmin,
                               float* __restrict__ tws, int N) {
  __shared__ float red[256];
  const int b = blockIdx.x, tid = threadIdx.x;
  float s = 0.f;
  for (int n = tid; n < N; n += 256)
    s += fsqrt_(fmaxf(nnmin[(size_t)b * N + n], EPSC));
  red[tid] = s; __syncthreads();
  for (int off = 128; off > 0; off >>= 1) {
    if (tid < off) red[tid] += red[tid + off];
    __syncthreads();
  }
  if (tid == 0) tws[b] = ALPHA_FACTOR * red[0] / (float)N;
}

// --------------------------------------------------------------- row min ---
__global__ void rowmin_kernel(const float* __restrict__ src,
                              const float* __restrict__ tgt,
                              float* __restrict__ rowmind, int N, int M) {
  __shared__ float tx_[2048], ty_[2048], tz_[2048];
  const int b = blockIdx.y;
  const float* S = src + (size_t)b * 3 * N;
  const float* T = tgt + (size_t)b * 3 * M;
  for (int i = threadIdx.x; i < M; i += blockDim.x) {
    tx_[i] = T[i]; ty_[i] = T[M + i]; tz_[i] = T[2 * M + i];
  }
  __syncthreads();
  const int wv = threadIdx.x >> 5, lane = threadIdx.x & 31;
  const int hf = lane >> 4, nl = lane & 15;
  const int n0 = (blockIdx.x * 8 + wv) * 16;
  const int ar = n0 + nl;
  v2f A;
  if (hf == 0) { A.x = -2.f * S[ar];        A.y = -2.f * S[N + ar]; }
  else         { A.x = -2.f * S[2 * N + ar]; A.y = 0.f; }
  float sn2[8];
#pragma unroll
  for (int v = 0; v < 8; ++v) {
    int r = n0 + v + 8 * hf;
    float x = S[r], y = S[N + r], z = S[2 * N + r];
    sn2[v] = x * x + y * y + z * z;
  }
  float mn[8];
#pragma unroll
  for (int v = 0; v < 8; ++v) mn[v] = BIGF;
  for (int m0 = 0; m0 < M; m0 += 16) {
    const int mc = m0 + nl;
    float tx = tx_[mc], ty = ty_[mc], tz = tz_[mc];
    v2f Bf;
    if (hf == 0) { Bf.x = tx; Bf.y = ty; } else { Bf.x = tz; Bf.y = 0.f; }
    const float tm2 = tx * tx + ty * ty + tz * tz;
    v8f C;
#pragma unroll
    for (int v = 0; v < 8; ++v) C[v] = sn2[v] + tm2;
    C = wmma16x16x4(A, Bf, C);
#pragma unroll
    for (int v = 0; v < 8; ++v) mn[v] = fminf(mn[v], C[v]);
  }
#pragma unroll
  for (int v = 0; v < 8; ++v) {
    float x = mn[v];
    for (int off = 8; off > 0; off >>= 1) x = fminf(x, __shfl_xor(x, off));
    if (nl == 0)
      rowmind[(size_t)b * N + n0 + v + 8 * hf] = fsqrt_(fmaxf(x, EPSC));
  }
}

// --------------------------------------------------------------- col min ---
__global__ void colmin_kernel(const float* __restrict__ src,
                              const float* __restrict__ tgt,
                              float* __restrict__ colmind, int N, int M) {
  __shared__ float sx[2048], sy[2048], sz[2048];
  const int b = blockIdx.y;
  const float* S = src + (size_t)b * 3 * N;
  const float* T = tgt + (size_t)b * 3 * M;
  for (int i = threadIdx.x; i < N; i += blockDim.x) {
    sx[i] = S[i]; sy[i] = S[N + i]; sz[i] = S[2 * N + i];
  }
  __syncthreads();
  const int wv = threadIdx.x >> 5, lane = threadIdx.x & 31;
  const int hf = lane >> 4, nl = lane & 15;
  const int m0 = (blockIdx.x * 8 + wv) * 16;
  const int mc = m0 + nl;
  float tx = T[mc], ty = T[M + mc], tz = T[2 * M + mc];
  v2f Bf;
  if (hf == 0) { Bf.x = tx; Bf.y = ty; } else { Bf.x = tz; Bf.y = 0.f; }
  const float tm2 = tx * tx + ty * ty + tz * tz;
  float mn = BIGF;
  for (int n0 = 0; n0 < N; n0 += 16) {
    const int ar = n0 + nl;
    v2f A;
    if (hf == 0) { A.x = -2.f * sx[ar]; A.y = -2.f * sy[ar]; }
    else         { A.x = -2.f * sz[ar]; A.y = 0.f; }
    v8f C;
#pragma unroll
    for (int v = 0; v < 8; ++v) {
      int r = n0 + v + 8 * hf;
      float x = sx[r], y = sy[r], z = sz[r];
      C[v] = x * x + y * y + z * z + tm2;
    }
    C = wmma16x16x4(A, Bf, C);
#pragma unroll
    for (int v = 0; v < 8; ++v) mn = fminf(mn, C[v]);
  }
  mn = fminf(mn, __shfl_xor(mn, 16));
  if (lane < 16) colmind[(size_t)b * M + mc] = fsqrt_(fmaxf(mn, EPSC));
}

// --------------------------------------------------------------- row sum ---
__global__ void rowsum_kernel(const float* __restrict__ src,
                              const float* __restrict__ tgt,
                              const float* __restrict__ tws,
                              const float* __restrict__ rowmind,
                              float* __restrict__ rowsum, int N, int M) {
  __shared__ float tx_[2048], ty_[2048], tz_[2048];
  const int b = blockIdx.y;
  const float* S = src + (size_t)b * 3 * N;
  const float* T = tgt + (size_t)b * 3 * M;
  for (int i = threadIdx.x; i < M; i += blockDim.x) {
    tx_[i] = T[i]; ty_[i] = T[M + i]; tz_[i] = T[2 * M + i];
  }
  __syncthreads();
  const float k = frcp_(tws[b]) * LOG2E;          // logits*log2e = -d*k
  const int wv = threadIdx.x >> 5, lane = threadIdx.x & 31;
  const int hf = lane >> 4, nl = lane & 15;
  const int n0 = (blockIdx.x * 8 + wv) * 16;
  const int ar = n0 + nl;
  v2f A;
  if (hf == 0) { A.x = -2.f * S[ar];        A.y = -2.f * S[N + ar]; }
  else         { A.x = -2.f * S[2 * N + ar]; A.y = 0.f; }
  float sn2[8], arow[8];
#pragma unroll
  for (int v = 0; v < 8; ++v) {
    int r = n0 + v + 8 * hf;
    float x = S[r], y = S[N + r], z = S[2 * N + r];
    sn2[v] = x * x + y * y + z * z;
    arow[v] = rowmind[(size_t)b * N + r] * k;     // rowmax*log2e (sign folded)
  }
  float acc[8] = {};
  for (int m0 = 0; m0 < M; m0 += 16) {
    const int mc = m0 + nl;
    float tx = tx_[mc], ty = ty_[mc], tz = tz_[mc];
    v2f Bf;
    if (hf == 0) { Bf.x = tx; Bf.y = ty; } else { Bf.x = tz; Bf.y = 0.f; }
    const float tm2 = tx * tx + ty * ty + tz * tz;
    v8f C;
#pragma unroll
    for (int v = 0; v < 8; ++v) C[v] = sn2[v] + tm2;
    C = wmma16x16x4(A, Bf, C);
#pragma unroll
    for (int v = 0; v < 8; ++v) {
      float d = fsqrt_(fmaxf(C[v], EPSC));
      acc[v] += fexp2_(fmaf(-d, k, arow[v]));     // exp((dmin-d)/t)
    }
  }
#pragma unroll
  for (int v = 0; v < 8; ++v) {
    float x = acc[v];
    for (int off = 8; off > 0; off >>= 1) x += __shfl_xor(x, off);
    if (nl == 0) rowsum[(size_t)b * N + n0 + v + 8 * hf] = x;
  }
}

// --------------------------------------------------------------- col sum ---
__global__ void colsum_kernel(const float* __restrict__ src,
                              const float* __restrict__ tgt,
                              const float* __restrict__ tws,
                              const float* __restrict__ colmind,
                              float* __restrict__ colsum, int N, int M) {
  __shared__ float sx[2048], sy[2048], sz[2048];
  const int b = blockIdx.y;
  const float* S = src + (size_t)b * 3 * N;
  const float* T = tgt + (size_t)b * 3 * M;
  for (int i = threadIdx.x; i < N; i += blockDim.x) {
    sx[i] = S[i]; sy[i] = S[N + i]; sz[i] = S[2 * N + i];
  }
  __syncthreads();
  const float k = frcp_(tws[b]) * LOG2E;
  const int wv = threadIdx.x >> 5, lane = threadIdx.x & 31;
  const int hf = lane >> 4, nl = lane & 15;
  const int m0 = (blockIdx.x * 8 + wv) * 16;
  const int mc = m0 + nl;
  float tx = T[mc], ty = T[M + mc], tz = T[2 * M + mc];
  v2f Bf;
  if (hf == 0) { Bf.x = tx; Bf.y = ty; } else { Bf.x = tz; Bf.y = 0.f; }
  const float tm2 = tx * tx + ty * ty + tz * tz;
  const float acol = colmind[(size_t)b * M + mc] * k;
  float acc = 0.f;
  for (int n0 = 0; n0 < N; n0 += 16) {
    const int ar = n0 + nl;
    v2f A;
    if (hf == 0) { A.x = -2.f * sx[ar]; A.y = -2.f * sy[ar]; }
    else         { A.x = -2.f * sz[ar]; A.y = 0.f; }
    v8f C;
#pragma unroll
    for (int v = 0; v < 8; ++v) {
      int r = n0 + v + 8 * hf;
      float x = sx[r], y = sy[r], z = sz[r];
      C[v] = x * x + y * y + z * z + tm2;
    }
    C = wmma16x16x4(A, Bf, C);
#pragma unroll
    for (int v = 0; v < 8; ++v) {
      float d = fsqrt_(fmaxf(C[v], EPSC));
      acc += fexp2_(fmaf(-d, k, acol));
    }
  }
  acc += __shfl_xor(acc, 16);
  if (lane < 16) colsum[(size_t)b * M + mc] = acc;
}

// ---------------------------------------------------------------- scores ---
__global__ void scores_kernel(const float* __restrict__ src,
                              const float* __restrict__ tgt,
                              const float* __restrict__ tws,
                              const float* __restrict__ rowmind,
                              const float* __restrict__ rowsum,
                              const float* __restrict__ colmind,
                              const float* __restrict__ colsum,
                              const int* __restrict__ iterp,
                              float* __restrict__ corr_out,
                              float* __restrict__ gamma, int N, int M) {
  __shared__ float tx_[2048], ty_[2048], tz_[2048];
  const int b = blockIdx.y;
  const float* S = src + (size_t)b * 3 * N;
  const float* T = tgt + (size_t)b * 3 * M;
  for (int i = threadIdx.x; i < M; i += blockDim.x) {
    tx_[i] = T[i]; ty_[i] = T[M + i]; tz_[i] = T[2 * M + i];
  }
  __syncthreads();
  const float k = frcp_(tws[b]) * LOG2E;
  const float twok = 2.0f * k;
  const int iter = iterp[0];
  const float kT = fexp2_((float)(iter - 1)) * LOG2E;   // (1/T)*log2e
  const int wv = threadIdx.x >> 5, lane = threadIdx.x & 31;
  const int hf = lane >> 4, nl = lane & 15;
  const int n0 = (blockIdx.x * 8 + wv) * 16;
  const int ar = n0 + nl;
  v2f A;
  if (hf == 0) { A.x = -2.f * S[ar];        A.y = -2.f * S[N + ar]; }
  else         { A.x = -2.f * S[2 * N + ar]; A.y = 0.f; }
  float sn2[8], arow[8], rsI[8];
#pragma unroll
  for (int v = 0; v < 8; ++v) {
    int r = n0 + v + 8 * hf;
    float x = S[r], y = S[N + r], z = S[2 * N + r];
    sn2[v] = x * x + y * y + z * z;
    arow[v] = rowmind[(size_t)b * N + r] * k;
    rsI[v] = frcp_(rowsum[(size_t)b * N + r]);
  }
  float aS[8] = {}, aG[8] = {}, aX[8] = {}, aY[8] = {}, aZ[8] = {};
  for (int m0 = 0; m0 < M; m0 += 16) {
    const int mc = m0 + nl;
    float tx = tx_[mc], ty = ty_[mc], tz = tz_[mc];
    v2f Bf;
    if (hf == 0) { Bf.x = tx; Bf.y = ty; } else { Bf.x = tz; Bf.y = 0.f; }
    const float tm2 = tx * tx + ty * ty + tz * tz;
    const float acol = colmind[(size_t)b * M + mc] * k;
    const float csI = frcp_(colsum[(size_t)b * M + mc]);
    v8f C;
#pragma unroll
    for (int v = 0; v < 8; ++v) C[v] = sn2[v] + tm2;
    C = wmma16x16x4(A, Bf, C);
#pragma unroll
    for (int v = 0; v < 8; ++v) {
      float d = fsqrt_(fmaxf(C[v], EPSC));
      // softmax_row * softmax_col folded into a single exp2
      float sc = fexp2_(fmaf(-d, twok, arow[v] + acol)) * (rsI[v] * csI);
      aS[v] += sc;
      aG[v] += sc * fexp2_(-d * kT);
      aX[v] += sc * tx;
      aY[v] += sc * ty;
      aZ[v] += sc * tz;
    }
  }
#pragma unroll
  for (int v = 0; v < 8; ++v) {
    for (int off = 8; off > 0; off >>= 1) {
      aS[v] += __shfl_xor(aS[v], off);
      aG[v] += __shfl_xor(aG[v], off);
      aX[v] += __shfl_xor(aX[v], off);
      aY[v] += __shfl_xor(aY[v], off);
      aZ[v] += __shfl_xor(aZ[v], off);
    }
    if (nl == 0) {
      int r = n0 + v + 8 * hf;
      float inv = 1.0f / (aS[v] + EPSC);
      corr_out[(size_t)b * 3 * N + r]         = aX[v] * inv;
      corr_out[(size_t)b * 3 * N + N + r]     = aY[v] * inv;
      corr_out[(size_t)b * 3 * N + 2 * N + r] = aZ[v] * inv;
      gamma[(size_t)b * N + r] = aG[v];
    }
  }
}

// -------------------------------------------------------------- finalize ---
__global__ void finalize_kernel(const float* __restrict__ src,
                                const float* __restrict__ corr,
                                const float* __restrict__ gamma,
                                float* __restrict__ Rout,
                                float* __restrict__ tout, int N) {
  const int b = blockIdx.x;
  const int tid = threadIdx.x;
  const float* S = src + (size_t)b * 3 * N;
  const float* Cc = corr + (size_t)b * 3 * N;
  const float* G = gamma + (size_t)b * N;
  __shared__ float red[256];

  auto reduce = [&](float v) -> float {
    red[tid] = v; __syncthreads();
    for (int off = 128; off > 0; off >>= 1) {
      if (tid < off) red[tid] += red[tid + off];
      __syncthreads();
    }
    float r = red[0]; __syncthreads();
    return r;
  };

  float pg = 0, psx = 0, psy = 0, psz = 0, pcx = 0, pcy = 0, pcz = 0;
  for (int n = tid; n < N; n += 256) {
    float g = G[n];
    pg += g;
    psx += g * S[n];  psy += g * S[N + n];  psz += g * S[2 * N + n];
    pcx += g * Cc[n]; pcy += g * Cc[N + n]; pcz += g * Cc[2 * N + n];
  }
  float sg = reduce(pg);
  float sx = reduce(psx), sy = reduce(psy), sz = reduce(psz);
  float cx = reduce(pcx), cy = reduce(pcy), cz = reduce(pcz);
  float gsum = sg + EPSC;
  float smx = sx / gsum, smy = sy / gsum, smz = sz / gsum;
  float cmx = cx / gsum, cmy = cy / gsum, cmz = cz / gsum;

  float h[9] = {};
  for (int n = tid; n < N; n += 256) {
    float g = G[n];
    float a0 = S[n] - smx, a1 = S[N + n] - smy, a2 = S[2 * N + n] - smz;
    float b0 = Cc[n] - cmx, b1 = Cc[N + n] - cmy, b2 = Cc[2 * N + n] - cmz;
    h[0] += g * a0 * b0; h[1] += g * a0 * b1; h[2] += g * a0 * b2;
    h[3] += g * a1 * b0; h[4] += g * a1 * b1; h[5] += g * a1 * b2;
    h[6] += g * a2 * b0; h[7] += g * a2 * b1; h[8] += g * a2 * b2;
  }
  float H[9];
  for (int k = 0; k < 9; ++k) H[k] = reduce(h[k]);
  if (tid != 0) return;

  H[0] += EPSC * 1.0f; H[4] += EPSC * 2.0f; H[8] += EPSC * 3.0f;

  float Amat[3][3];
  for (int i = 0; i < 3; ++i)
    for (int j = 0; j < 3; ++j) {
      float acc = 0.f;
      for (int kk = 0; kk < 3; ++kk) acc += H[kk * 3 + i] * H[kk * 3 + j];
      Amat[i][j] = acc;
    }
  float V[3][3] = {{1, 0, 0}, {0, 1, 0}, {0, 0, 1}};
  for (int sweep = 0; sweep < 25; ++sweep) {
    for (int p = 0; p < 2; ++p)
      for (int q = p + 1; q < 3; ++q) {
        float apq = Amat[p][q];
        if (fabsf(apq) < 1e-24f) continue;
        float tau = (Amat[q][q] - Amat[p][p]) / (2.0f * apq);
        float tt = (tau >= 0.f ? 1.f : -1.f) / (fabsf(tau) + sqrtf(1.f + tau * tau));
        float c = 1.0f / sqrtf(1.f + tt * tt);
        float s = tt * c;
        for (int kk = 0; kk < 3; ++kk) {
          float akp = Amat[kk][p], akq = Amat[kk][q];
          Amat[kk][p] = c * akp - s * akq;
          Amat[kk][q] = s * akp + c * akq;
        }
        for (int kk = 0; kk < 3; ++kk) {
          float apk = Amat[p][kk], aqk = Amat[q][kk];
          Amat[p][kk] = c * apk - s * aqk;
          Amat[q][kk] = s * apk + c * aqk;
        }
        for (int kk = 0; kk < 3; ++kk) {
          float vkp = V[kk][p], vkq = V[kk][q];
          V[kk][p] = c * vkp - s * vkq;
          V[kk][q] = s * vkp + c * vkq;
        }
      }
  }
  float ev[3] = {Amat[0][0], Amat[1][1], Amat[2][2]};
  for (int i = 0; i < 2; ++i)
    for (int j = i + 1; j < 3; ++j)
      if (ev[j] > ev[i]) {
        float t0 = ev[i]; ev[i] = ev[j]; ev[j] = t0;
        for (int kk = 0; kk < 3; ++kk) {
          float t1 = V[kk][i]; V[kk][i] = V[kk][j]; V[kk][j] = t1;
        }
      }
  float U[3][3], hv[3];
  for (int i = 0; i < 3; ++i)
    hv[i] = H[i * 3] * V[0][0] + H[i * 3 + 1] * V[1][0] + H[i * 3 + 2] * V[2][0];
  float nu0 = sqrtf(hv[0] * hv[0] + hv[1] * hv[1] + hv[2] * hv[2]) + 1e-30f;
  for (int i = 0; i < 3; ++i) U[i][0] = hv[i] / nu0;
  for (int i = 0; i < 3; ++i)
    hv[i] = H[i * 3] * V[0][1] + H[i * 3 + 1] * V[1][1] + H[i * 3 + 2] * V[2][1];
  float dp = U[0][0] * hv[0] + U[1][0] * hv[1] + U[2][0] * hv[2];
  for (int i = 0; i < 3; ++i) hv[i] -= dp * U[i][0];
  float nu1 = sqrtf(hv[0] * hv[0] + hv[1] * hv[1] + hv[2] * hv[2]) + 1e-30f;
  for (int i = 0; i < 3; ++i) U[i][1] = hv[i] / nu1;
  U[0][2] = U[1][0] * U[2][1] - U[2][0] * U[1][1];
  U[1][2] = U[2][0] * U[0][1] - U[0][0] * U[2][1];
  U[2][2] = U[0][0] * U[1][1] - U[1][0] * U[0][1];

  float R[3][3];
  for (int i = 0; i < 3; ++i)
    for (int j = 0; j < 3; ++j)
      R[i][j] = V[i][0] * U[j][0] + V[i][1] * U[j][1] + V[i][2] * U[j][2];
  float det = R[0][0] * (R[1][1] * R[2][2] - R[1][2] * R[2][1])
            - R[0][1] * (R[1][0] * R[2][2] - R[1][2] * R[2][0])
            + R[0][2] * (R[1][0] * R[2][1] - R[1][1] * R[2][0]);
  if (det < 0.f)
    for (int i = 0; i < 3; ++i)
      for (int j = 0; j < 3; ++j) R[i][j] -= 2.0f * V[i][2] * U[j][2];

  for (int i = 0; i < 3; ++i)
    for (int j = 0; j < 3; ++j) Rout[b * 9 + i * 3 + j] = R[i][j];
  float sm[3] = {smx, smy, smz}, cmv[3] = {cmx, cmy, cmz};
  for (int i = 0; i < 3; ++i)
    tout[b * 3 + i] = cmv[i] - (R[i][0] * sm[0] + R[i][1] * sm[1] + R[i][2] * sm[2]);
}

// ---------------------------------------------------------------------------
extern "C" void kernel_launch(void* const* d_in, const int* in_sizes, int n_in,
                              void* d_out, int out_size, void* d_ws, size_t ws_size,
                              hipStream_t stream) {
  (void)in_sizes; (void)n_in; (void)out_size; (void)ws_size;
  const float* src = (const float*)d_in[0];   // [B,3,N]
  const float* tgt = (const float*)d_in[1];   // [B,3,M]
  const int* iterp = (const int*)d_in[2];     // scalar
  const int B = 8, N = 2048, M = 2048;        // per reference setup_inputs()

  float* out = (float*)d_out;
  float* Rout = out;                  // B*9
  float* tout = out + B * 9;          // B*3
  float* corr = out + B * 9 + B * 3;  // B*3*N

  float* ws = (float*)d_ws;
  float* tws     = ws;                          // B
  float* nnmin   = tws + B;                     // B*N
  float* rowmind = nnmin + (size_t)B * N;       // B*N
  float* rowsum  = rowmind + (size_t)B * N;     // B*N
  float* colmind = rowsum + (size_t)B * N;      // B*M
  float* colsum  = colmind + (size_t)B * M;     // B*M
  float* gamma   = colsum + (size_t)B * M;      // B*N

  selfmin_kernel<<<dim3(N / 128, B), 256, 0, stream>>>(src, nnmin, N);
  treduce_kernel<<<B, 256, 0, stream>>>(nnmin, tws, N);
  rowmin_kernel<<<dim3(N / 128, B), 256, 0, stream>>>(src, tgt, rowmind, N, M);
  colmin_kernel<<<dim3(M / 128, B), 256, 0, stream>>>(src, tgt, colmind, N, M);
  rowsum_kernel<<<dim3(N / 128, B), 256, 0, stream>>>(src, tgt, tws, rowmind,
                                                      rowsum, N, M);
  colsum_kernel<<<dim3(M / 128, B), 256, 0, stream>>>(src, tgt, tws, colmind,
                                                      colsum, N, M);
  scores_kernel<<<dim3(N / 128, B), 256, 0, stream>>>(src, tgt, tws, rowmind,
                                                      rowsum, colmind, colsum,
                                                      iterp, corr, gamma, N, M);
  finalize_kernel<<<B, 256, 0, stream>>>(src, corr, gamma, Rout, tout, N);
}